// GCN_257698038541
// MI455X (gfx1250) — compile-verified
//
#include <hip/hip_runtime.h>
#include <math.h>

typedef __attribute__((ext_vector_type(2))) float v2f;
typedef __attribute__((ext_vector_type(8))) float v8f;

// ---------------------------------------------------------------------------
// Weight transpose with column padding: WT[n*K + k] = (n < N) ? W[k*N + n] : 0
// ---------------------------------------------------------------------------
__global__ void k_transpose_pad(const float* __restrict__ W, float* __restrict__ WT,
                                int K, int N, int Npad) {
    int idx = blockIdx.x * 256 + threadIdx.x;      // over Npad*K
    if (idx >= Npad * K) return;
    int n = idx / K;
    int k = idx - n * K;
    WT[n * K + k] = (n < N) ? W[k * N + n] : 0.0f;
}

__global__ void k_padvec(const float* __restrict__ b, float* __restrict__ bp,
                         int n, int npad) {
    int i = blockIdx.x * 64 + threadIdx.x;
    if (i < npad) bp[i] = (i < n) ? b[i] : 0.0f;
}

// ---------------------------------------------------------------------------
// Edge scatter-sum: out[dst[e]] += x[src[e]]  (256-wide rows, float4 per thread,
// 64 threads per edge). Hardware f32 atomics -> L2 atomic units; the 51 MB
// feature matrix is L2-resident on MI455X (192 MB L2).
// ---------------------------------------------------------------------------
__global__ void __launch_bounds__(256)
k_scatter(const float* __restrict__ x, const int* __restrict__ src,
          const int* __restrict__ dst, float* __restrict__ out, int E) {
    int e = blockIdx.x * 4 + (threadIdx.x >> 6);
    if (e >= E) return;
    int c = (threadIdx.x & 63) * 4;
    int s = src[e];
    int d = dst[e];
    const float4 v = *(const float4*)(x + (size_t)s * 256 + c);
    float* o = out + (size_t)d * 256 + c;
    unsafeAtomicAdd(o + 0, v.x);
    unsafeAtomicAdd(o + 1, v.y);
    unsafeAtomicAdd(o + 2, v.z);
    unsafeAtomicAdd(o + 3, v.w);
}

// ---------------------------------------------------------------------------
// C[M x LDC] = act(A[M x 256] @ WT^T + bias) via V_WMMA_F32_16X16X4_F32.
// One 16x16 output tile per wave; K looped in steps of 4 (64 WMMAs/tile).
// WT is [Npad x 256] (pre-transposed) so both operands are contiguous v2f
// loads with the same addressing pattern. LDC/RELU are compile-time so the
// epilogue stores fold row offsets into the instruction offset field.
// ---------------------------------------------------------------------------
template <int LDC, bool RELU>
__global__ void __launch_bounds__(256)
k_gemm_wmma(const float* __restrict__ A, const float* __restrict__ WT,
            const float* __restrict__ bias, float* __restrict__ C,
            int Mtiles, int Ntiles, int Mrows) {
    const int wave = threadIdx.x >> 5;
    const int lane = threadIdx.x & 31;
    const int tile = blockIdx.x * 8 + wave;
    if (tile >= Mtiles * Ntiles) return;          // wave-uniform: EXEC stays full
    const int mt = tile / Ntiles;
    const int nt = tile - mt * Ntiles;

    const int half = lane >> 4;                   // 0: K={0,1}, 1: K={2,3}
    const int l15  = lane & 15;

    int arow = mt * 16 + l15;
    if (arow >= Mrows) arow = Mrows - 1;          // clamp keeps EXEC all-ones
    const float* ap = A  + (size_t)arow * 256 + half * 2;
    const float* bp = WT + (size_t)(nt * 16 + l15) * 256 + half * 2;

    v8f acc = {};
#pragma unroll 8
    for (int k = 0; k < 256; k += 4) {
        v2f a = *(const v2f*)(ap + k);
        v2f b = *(const v2f*)(bp + k);
        // (neg_a, A, neg_b, B, c_mod, C, reuse_a, reuse_b)
        acc = __builtin_amdgcn_wmma_f32_16x16x4_f32(
            false, a, false, b, (short)0, acc, false, false);
    }

    const int col  = nt * 16 + l15;
    const float bv = bias[col];
    const int row0 = mt * 16 + half * 8;          // D: VGPR r -> row0 + r
    float* cp = C + (size_t)row0 * LDC + col;

    if (mt * 16 + 16 <= Mrows) {                  // wave-uniform full-tile path
#pragma unroll
        for (int r = 0; r < 8; ++r) {
            float v = acc[r] + bv;
            if (RELU) v = fmaxf(v, 0.0f);
            cp[r * LDC] = v;                      // constant byte offsets
        }
    } else {                                      // ragged M tail (unused for N=50000)
#pragma unroll
        for (int r = 0; r < 8; ++r) {
            if (row0 + r < Mrows) {
                float v = acc[r] + bv;
                if (RELU) v = fmaxf(v, 0.0f);
                cp[r * LDC] = v;
            }
        }
    }
}

// ---------------------------------------------------------------------------
// Row-wise log_softmax over 40 valid cols of a 48-col padded buffer.
// One wave32 per row; lane covers cols {lane, lane+32}.
// ---------------------------------------------------------------------------
__global__ void __launch_bounds__(256)
k_logsoftmax(const float* __restrict__ logits, float* __restrict__ out, int Nrows) {
    const int wave = threadIdx.x >> 5;
    const int lane = threadIdx.x & 31;
    const int row  = blockIdx.x * 8 + wave;
    if (row >= Nrows) return;
    const float* p = logits + (size_t)row * 48;
    float x0 = p[lane];                                   // lane < 32 < 40: valid
    bool  h1 = (lane + 32) < 40;
    float x1 = h1 ? p[lane + 32] : -INFINITY;

    float m = fmaxf(x0, x1);
    for (int off = 16; off > 0; off >>= 1)
        m = fmaxf(m, __shfl_xor(m, off, 32));
    float s = __expf(x0 - m) + (h1 ? __expf(x1 - m) : 0.0f);
    for (int off = 16; off > 0; off >>= 1)
        s += __shfl_xor(s, off, 32);
    float L = m + __logf(s);

    float* o = out + (size_t)row * 40;
    o[lane] = x0 - L;
    if (h1) o[lane + 32] = x1 - L;
}

__global__ void k_scalar(float* out, float v) { out[0] = v; }

// ---------------------------------------------------------------------------
extern "C" void kernel_launch(void* const* d_in, const int* in_sizes, int n_in,
                              void* d_out, int out_size, void* d_ws, size_t ws_size,
                              hipStream_t stream) {
    const float* features = (const float*)d_in[0];
    const int*   edge     = (const int*)d_in[1];
    const float* W1 = (const float*)d_in[4];
    const float* b1 = (const float*)d_in[5];
    const float* W2 = (const float*)d_in[6];
    const float* b2 = (const float*)d_in[7];
    const float* W3 = (const float*)d_in[8];
    const float* b3 = (const float*)d_in[9];
    float* out = (float*)d_out;

    const int Nn = in_sizes[0] / 256;   // 50000 nodes
    const int E  = in_sizes[1] / 2;     // 800000 edges
    const int* src = edge;
    const int* dst = edge + E;

    float* ws     = (float*)d_ws;
    float* bufA   = ws;                               // [Nn x 256] aggregation
    float* bufB   = bufA + (size_t)Nn * 256;          // [Nn x 256] hidden
    float* logits = bufB + (size_t)Nn * 256;          // [Nn x 48]
    float* w1t    = logits + (size_t)Nn * 48;         // [256 x 256]
    float* w2t    = w1t + 256 * 256;                  // [256 x 256]
    float* w3t    = w2t + 256 * 256;                  // [48  x 256]
    float* b3p    = w3t + 48 * 256;                   // [48]

    const size_t aggBytes = (size_t)Nn * 256 * sizeof(float);
    const int Mtiles  = (Nn + 15) / 16;
    const int sBlocks = (E + 3) / 4;
    const int gBlocksH = (Mtiles * 16 + 7) / 8;       // hidden layers, 16 n-tiles
    const int gBlocksO = (Mtiles * 3 + 7) / 8;        // output layer, 3 n-tiles

    // Weight prep (transpose + pad)
    k_transpose_pad<<<(256 * 256 + 255) / 256, 256, 0, stream>>>(W1, w1t, 256, 256, 256);
    k_transpose_pad<<<(256 * 256 + 255) / 256, 256, 0, stream>>>(W2, w2t, 256, 256, 256);
    k_transpose_pad<<<( 48 * 256 + 255) / 256, 256, 0, stream>>>(W3, w3t, 256,  40,  48);
    k_padvec<<<1, 64, 0, stream>>>(b3, b3p, 40, 48);

    // Layer 1: agg(features) -> relu(@W1 + b1)
    hipMemsetAsync(bufA, 0, aggBytes, stream);
    k_scatter<<<sBlocks, 256, 0, stream>>>(features, src, dst, bufA, E);
    k_gemm_wmma<256, true><<<gBlocksH, 256, 0, stream>>>(bufA, w1t, b1, bufB,
                                                         Mtiles, 16, Nn);
    // Layer 2: agg(h1) -> relu(@W2 + b2)
    hipMemsetAsync(bufA, 0, aggBytes, stream);
    k_scatter<<<sBlocks, 256, 0, stream>>>(bufB, src, dst, bufA, E);
    k_gemm_wmma<256, true><<<gBlocksH, 256, 0, stream>>>(bufA, w2t, b2, bufB,
                                                         Mtiles, 16, Nn);
    // Output layer: agg(h2) -> @W3 + b3 (48-col padded), then log_softmax
    hipMemsetAsync(bufA, 0, aggBytes, stream);
    k_scatter<<<sBlocks, 256, 0, stream>>>(bufB, src, dst, bufA, E);
    k_gemm_wmma<48, false><<<gBlocksO, 256, 0, stream>>>(bufA, w3t, b3p, logits,
                                                         Mtiles, 3, Nn);
    k_logsoftmax<<<(Nn + 7) / 8, 256, 0, stream>>>(logits, out, Nn);

    // node_count = (N_LAYERS + 1) * N, appended if the harness expects it
    if (out_size > Nn * 40)
        k_scalar<<<1, 1, 0, stream>>>(out + (size_t)Nn * 40, 3.0f * (float)Nn);
}